// QModel_65481071408982
// MI455X (gfx1250) — compile-verified
//
#include <hip/hip_runtime.h>
#include <hip/hip_bf16.h>
#include <math.h>

// ---------------------------------------------------------------------------
// 12-qubit circuit sim as complex 64x64 GEMMs on V_WMMA_F32_16X16X4_F32.
// State S[r,c]: r = wires 0..5 (wire0 = MSB of r), c = wires 6..11.
// Per layer:   S <- A @ S          (A = rowCNOTperm * Kron(U0..U5))
//              S <- S @ C^T        (C = Kron(U6..U11))
//              permuted writeback  (CNOT(5,6) straddle + column CNOT chain)
// Head folded: out[b] = sum_{r,c} |S|^2 * (fr[r] + gc[c]) + head_b.
// ---------------------------------------------------------------------------

typedef float v2f __attribute__((ext_vector_type(2)));
typedef float v8f __attribute__((ext_vector_type(8)));

#define SROW 68          // padded LDS row stride (even -> 8B-aligned b64 loads)
#define NLAY 2

struct cf { float re, im; };
__device__ __forceinline__ cf cmul(cf a, cf b) {
    return { a.re * b.re - a.im * b.im, a.re * b.im + a.im * b.re };
}

// ---------------------------------------------------------------------------
// Setup kernel: build per-layer operand matrices in WMMA fragment order.
//   Aop[l][plane][4096]  plane: 0=re 1=im 2=-im   (A-operand layout, left GEMM)
//   Bop[l][plane][4096]  value C[N,K] = (C^T)[K,N] (B-operand layout, right GEMM)
//   frgc[128]: fr[64] then gc[64]  (sign-weighted head vectors)
// Slot decode: slot = ((tile*16 + s)*32 + lane)*2 + v
//   A-op: M = tile*16 + lane%16, K = s*4 + (lane/16)*2 + v   (ISA 16x4 f32 A layout)
//   B-op: N = tile*16 + lane%16, K = s*4 + (lane/16)*2 + v
// ---------------------------------------------------------------------------
__global__ __launch_bounds__(128) void qprep_kernel(
    const float* __restrict__ params,   // (NLAY, 12, 3)
    const float* __restrict__ head_w,   // (12)
    float* __restrict__ Aop,            // (NLAY, 3, 4096)
    float* __restrict__ Bop,            // (NLAY, 3, 4096)
    float* __restrict__ frgc)           // (128)
{
    __shared__ cf U[12][2][2];
    const int tid = threadIdx.x;

    for (int l = 0; l < NLAY; ++l) {
        if (tid < 12) {
            const float pz = params[(l * 12 + tid) * 3 + 0];
            const float px = params[(l * 12 + tid) * 3 + 1];
            const float py = params[(l * 12 + tid) * 3 + 2];
            const float tz = 0.5f * pz, tx = 0.5f * px, ty = 0.5f * py;
            const cf ez  = { cosf(tz), -sinf(tz) };   // e^{-i tz}
            const cf ezc = { ez.re, -ez.im };
            const float cx = cosf(tx), sx = sinf(tx);
            const float cy = cosf(ty), sy = sinf(ty);
            // M = RX*RZ :  -i*sx*e = { sx*e.im, -sx*e.re }
            const cf M00 = { cx * ez.re,  cx * ez.im  };
            const cf M01 = { sx * ezc.im, -sx * ezc.re };
            const cf M10 = { sx * ez.im,  -sx * ez.re  };
            const cf M11 = { cx * ezc.re, cx * ezc.im };
            // U = RY * M
            U[tid][0][0] = { cy * M00.re - sy * M10.re, cy * M00.im - sy * M10.im };
            U[tid][0][1] = { cy * M01.re - sy * M11.re, cy * M01.im - sy * M11.im };
            U[tid][1][0] = { sy * M00.re + cy * M10.re, sy * M00.im + cy * M10.im };
            U[tid][1][1] = { sy * M01.re + cy * M11.re, sy * M01.im + cy * M11.im };
        }
        __syncthreads();

        for (int slot = tid; slot < 4096; slot += 128) {
            const int v    = slot & 1;
            const int lane = (slot >> 1) & 31;
            const int s    = (slot >> 6) & 15;
            const int tile = slot >> 10;
            const int MN   = tile * 16 + (lane & 15);
            const int K    = s * 4 + (lane >> 4) * 2 + v;

            // A[M,K] = R6[chainInv(M), K]; row-CNOT-chain inverse = M ^ (M>>1)
            const int j = (MN ^ (MN >> 1)) & 63;
            cf av = { 1.f, 0.f };
            cf bv = { 1.f, 0.f };
            #pragma unroll
            for (int w = 0; w < 6; ++w) {
                av = cmul(av, U[w][(j >> (5 - w)) & 1][(K >> (5 - w)) & 1]);
                bv = cmul(bv, U[6 + w][(MN >> (5 - w)) & 1][(K >> (5 - w)) & 1]);
            }
            Aop[(l * 3 + 0) * 4096 + slot] = av.re;
            Aop[(l * 3 + 1) * 4096 + slot] = av.im;
            Aop[(l * 3 + 2) * 4096 + slot] = -av.im;
            Bop[(l * 3 + 0) * 4096 + slot] = bv.re;
            Bop[(l * 3 + 1) * 4096 + slot] = bv.im;
            Bop[(l * 3 + 2) * 4096 + slot] = -bv.im;
        }
        __syncthreads();   // protect U before next layer overwrites it
    }

    if (tid < 64) {
        float f = 0.f;
        #pragma unroll
        for (int w = 0; w < 6; ++w)
            f += (((tid >> (5 - w)) & 1) ? -1.f : 1.f) * head_w[w];
        frgc[tid] = f;
    } else if (tid < 128) {
        const int c = tid - 64;
        float g = 0.f;
        #pragma unroll
        for (int w = 0; w < 6; ++w)
            g += (((c >> (5 - w)) & 1) ? -1.f : 1.f) * head_w[6 + w];
        frgc[tid] = g;
    }
}

// ---------------------------------------------------------------------------
// Main kernel: one sample per block, 128 threads = 4 waves. Wave w owns output
// rows [16w, 16w+16). State lives in LDS; GEMMs accumulate in v8f registers.
// ---------------------------------------------------------------------------
__global__ __launch_bounds__(128) void qsim_kernel(
    const float* __restrict__ x,      // (B, 12)
    const float* __restrict__ head_b, // (1)
    const float* __restrict__ Aop,    // (NLAY, 3, 4096)
    const float* __restrict__ Bop,    // (NLAY, 3, 4096)
    const float* __restrict__ frgc,   // (128)
    float* __restrict__ out)          // (B)
{
    __shared__ float sRe[64 * SROW];
    __shared__ float sIm[64 * SROW];
    __shared__ float ca[64], cb[64], fr[64], gc[64];
    __shared__ float red[128];

    const int b     = blockIdx.x;
    const int tid   = threadIdx.x;
    const int lane  = tid & 31;
    const int wave  = tid >> 5;
    const int laneM = lane & 15;
    const int laneH = lane >> 4;

    // --- encoding: S0 = a ⊗ b (real, rank-1), plus fr/gc load ---
    if (tid < 64) {
        fr[tid] = frgc[tid];
        float prod = 1.f;
        #pragma unroll
        for (int w = 0; w < 6; ++w) {
            const float t = 0.5f * x[b * 12 + w];
            prod *= (((tid >> (5 - w)) & 1) ? sinf(t) : cosf(t));
        }
        ca[tid] = prod;
    } else {
        const int c = tid - 64;
        gc[c] = frgc[tid];
        float prod = 1.f;
        #pragma unroll
        for (int w = 0; w < 6; ++w) {
            const float t = 0.5f * x[b * 12 + 6 + w];
            prod *= (((c >> (5 - w)) & 1) ? sinf(t) : cosf(t));
        }
        cb[c] = prod;
    }
    __syncthreads();
    for (int e = tid; e < 4096; e += 128) {
        const int r = e >> 6, c = e & 63;
        sRe[r * SROW + c] = ca[r] * cb[c];
        sIm[r * SROW + c] = 0.f;
    }
    __syncthreads();

    for (int l = 0; l < NLAY; ++l) {
        const float* Ar = Aop + (l * 3 + 0) * 4096;
        const float* Ai = Aop + (l * 3 + 1) * 4096;
        const float* An = Aop + (l * 3 + 2) * 4096;

        // ================= LEFT: D = A @ S (complex) =================
        v8f accRe[4], accIm[4];
        #pragma unroll
        for (int nt = 0; nt < 4; ++nt) {
            accRe[nt] = (v8f){0.f, 0.f, 0.f, 0.f, 0.f, 0.f, 0.f, 0.f};
            accIm[nt] = (v8f){0.f, 0.f, 0.f, 0.f, 0.f, 0.f, 0.f, 0.f};
        }
        for (int s = 0; s < 16; ++s) {
            const int aidx = ((wave * 16 + s) * 32 + lane) * 2;
            const v2f ar = *(const v2f*)(Ar + aidx);
            const v2f ai = *(const v2f*)(Ai + aidx);
            const v2f an = *(const v2f*)(An + aidx);
            const int k0 = s * 4 + laneH * 2;
            #pragma unroll
            for (int nt = 0; nt < 4; ++nt) {
                const int col = nt * 16 + laneM;
                v2f br, bi;
                br.x = sRe[(k0 + 0) * SROW + col];
                br.y = sRe[(k0 + 1) * SROW + col];
                bi.x = sIm[(k0 + 0) * SROW + col];
                bi.y = sIm[(k0 + 1) * SROW + col];
                // re += Ar*Sr + (-Ai)*Si ; im += Ar*Si + Ai*Sr
                accRe[nt] = __builtin_amdgcn_wmma_f32_16x16x4_f32(false, ar, false, br, (short)0, accRe[nt], false, false);
                accRe[nt] = __builtin_amdgcn_wmma_f32_16x16x4_f32(false, an, false, bi, (short)0, accRe[nt], false, false);
                accIm[nt] = __builtin_amdgcn_wmma_f32_16x16x4_f32(false, ar, false, bi, (short)0, accIm[nt], false, false);
                accIm[nt] = __builtin_amdgcn_wmma_f32_16x16x4_f32(false, ai, false, br, (short)0, accIm[nt], false, false);
            }
        }
        __syncthreads();   // all waves finished reading S
        #pragma unroll
        for (int nt = 0; nt < 4; ++nt)
            #pragma unroll
            for (int v = 0; v < 8; ++v) {
                const int r = wave * 16 + v + laneH * 8;
                const int c = nt * 16 + laneM;
                sRe[r * SROW + c] = accRe[nt][v];
                sIm[r * SROW + c] = accIm[nt][v];
            }
        __syncthreads();

        // ================= RIGHT: D = S @ C^T (complex) =================
        const float* Br = Bop + (l * 3 + 0) * 4096;
        const float* Bi = Bop + (l * 3 + 1) * 4096;
        const float* Bn = Bop + (l * 3 + 2) * 4096;
        #pragma unroll
        for (int nt = 0; nt < 4; ++nt) {
            accRe[nt] = (v8f){0.f, 0.f, 0.f, 0.f, 0.f, 0.f, 0.f, 0.f};
            accIm[nt] = (v8f){0.f, 0.f, 0.f, 0.f, 0.f, 0.f, 0.f, 0.f};
        }
        for (int s = 0; s < 16; ++s) {
            const int m  = wave * 16 + laneM;
            const int k0 = s * 4 + laneH * 2;
            const v2f arS = *(const v2f*)(&sRe[m * SROW + k0]);   // ds_load_b64
            const v2f aiS = *(const v2f*)(&sIm[m * SROW + k0]);
            #pragma unroll
            for (int nt = 0; nt < 4; ++nt) {
                const int bidx = ((nt * 16 + s) * 32 + lane) * 2;
                const v2f br = *(const v2f*)(Br + bidx);
                const v2f bi = *(const v2f*)(Bi + bidx);
                const v2f bn = *(const v2f*)(Bn + bidx);
                // re += Sr*Cr + Si*(-Ci) ; im += Sr*Ci + Si*Cr
                accRe[nt] = __builtin_amdgcn_wmma_f32_16x16x4_f32(false, arS, false, br, (short)0, accRe[nt], false, false);
                accRe[nt] = __builtin_amdgcn_wmma_f32_16x16x4_f32(false, aiS, false, bn, (short)0, accRe[nt], false, false);
                accIm[nt] = __builtin_amdgcn_wmma_f32_16x16x4_f32(false, arS, false, bi, (short)0, accIm[nt], false, false);
                accIm[nt] = __builtin_amdgcn_wmma_f32_16x16x4_f32(false, aiS, false, br, (short)0, accIm[nt], false, false);
            }
        }
        __syncthreads();
        // permuted writeback: CNOT(5,6) straddle then column CNOT chain
        #pragma unroll
        for (int nt = 0; nt < 4; ++nt)
            #pragma unroll
            for (int v = 0; v < 8; ++v) {
                const int r  = wave * 16 + v + laneH * 8;
                const int c  = nt * 16 + laneM;
                const int cx = c ^ ((v & 1) << 5);                   // (r&1) == (v&1)
                const int cd = (cx ^ (cx >> 1) ^ (cx >> 2) ^ (cx >> 3) ^ (cx >> 4) ^ (cx >> 5)) & 63;
                sRe[r * SROW + cd] = accRe[nt][v];
                sIm[r * SROW + cd] = accIm[nt][v];
            }
        __syncthreads();
    }

    // --- fused probs + signs + head: out = sum |S|^2 (fr[r]+gc[c]) + head_b ---
    float acc = 0.f;
    for (int e = tid; e < 4096; e += 128) {
        const int r = e >> 6, c = e & 63;
        const float re = sRe[r * SROW + c];
        const float im = sIm[r * SROW + c];
        acc += (re * re + im * im) * (fr[r] + gc[c]);
    }
    red[tid] = acc;
    __syncthreads();
    if (tid < 32) {
        float v = red[tid] + red[tid + 32] + red[tid + 64] + red[tid + 96];
        #pragma unroll
        for (int off = 16; off > 0; off >>= 1)
            v += __shfl_down(v, off, 32);
        if (tid == 0) out[b] = v + head_b[0];
    }
}

// ---------------------------------------------------------------------------
extern "C" void kernel_launch(void* const* d_in, const int* in_sizes, int n_in,
                              void* d_out, int out_size, void* d_ws, size_t ws_size,
                              hipStream_t stream) {
    (void)n_in; (void)out_size; (void)ws_size;
    const float* x      = (const float*)d_in[0];   // (B, 12)
    const float* params = (const float*)d_in[1];   // (2, 12, 3)
    const float* head_w = (const float*)d_in[2];   // (12)
    const float* head_b = (const float*)d_in[3];   // (1)
    float* out = (float*)d_out;

    float* ws   = (float*)d_ws;
    float* Aop  = ws;                       // NLAY*3*4096 floats
    float* Bop  = Aop + NLAY * 3 * 4096;    // NLAY*3*4096 floats
    float* frgc = Bop + NLAY * 3 * 4096;    // 128 floats

    const int B = in_sizes[0] / 12;

    qprep_kernel<<<1, 128, 0, stream>>>(params, head_w, Aop, Bop, frgc);
    qsim_kernel<<<B, 128, 0, stream>>>(x, head_b, Aop, Bop, frgc, out);
}